// UnitaryLinear_88630945120606
// MI455X (gfx1250) — compile-verified
//
#include <hip/hip_runtime.h>

// ---------------------------------------------------------------------------
// UnitaryLinear (Cayley transform) for MI455X / gfx1250, wave32 + WMMA fp32.
//   A = W - W^H (skew-Hermitian), M = I - A, P = I + A = M^H
//   M^{-1} via Newton-Schulz: X_{k+1} = X_k (2I - M X_k), X_0 = alpha * P
//   (Hermitian part of M is exactly I -> cond(M) <= ~3, quadratic convergence)
//   U = M^{-1} P ;  y = x U^T  (complex64 interleaved output)
// Everything GEMM-shaped -> v_wmma_f32_16x16x4_f32 throughout (exact fp32).
//
// Round-2 changes (from asm inspection):
//  * LDS layouts chosen so every WMMA fragment is a single aligned ds_load_b64
//    into an even VGPR pair (A stored M-major, B stored N-major, row stride
//    BK+4=36 -> conflict-free b64 reads, 16B-aligned b128 fills).
//  * 3 accumulators per tile (re+, re-, im): no per-step VALU negation.
//  * #pragma unroll 4 on the k-step loop: stay under 256 VGPRs, no
//    s_set_vgpr_msb thrash.
// ---------------------------------------------------------------------------

typedef float v2f __attribute__((ext_vector_type(2)));
typedef float v8f __attribute__((ext_vector_type(8)));

static __device__ __forceinline__ v8f wmma_f32(v2f a, v2f b, v8f c) {
  // (neg_a, A, neg_b, B, c_mod, C, reuse_a, reuse_b)
  return __builtin_amdgcn_wmma_f32_16x16x4_f32(false, a, false, b, (short)0, c,
                                               false, false);
}

#define BM 64
#define BN 64
#define BK 32
#define LSTR (BK + 4)  // 36: conflict-free b64 frag reads, 16B-aligned fills

// Complex GEMM, C = A * B, all row-major, dims multiples of 64/32.
// A,B,C stored as separate re/im fp32 planes; optional interleaved C output.
__launch_bounds__(128)
__global__ void cgemm_nn(const float* __restrict__ Ar, const float* __restrict__ Ai,
                         const float* __restrict__ Br, const float* __restrict__ Bi,
                         float* __restrict__ Cr, float* __restrict__ Ci,
                         float* __restrict__ Cint,
                         int Mdim, int Ndim, int Kdim, int interleave) {
  __shared__ float sAr[BM][LSTR];  // A tile, M-major: frag = b64 at [m][ka]
  __shared__ float sAi[BM][LSTR];
  __shared__ float sBr[BN][LSTR];  // B tile, N-major: frag = b64 at [n][ka]
  __shared__ float sBi[BN][LSTR];

  const int t    = threadIdx.x;
  const int wave = t >> 5;
  const int lane = t & 31;
  const int half = lane >> 4;   // 0: K pair {0,1}, 1: K pair {2,3} (ISA layout)
  const int l16  = lane & 15;
  const int bm   = blockIdx.y * BM;
  const int bn   = blockIdx.x * BN;
  const int wm   = (wave >> 1) * 32;  // 2x2 waves -> 32x32 wave tile
  const int wn   = (wave & 1) * 32;

  // 3 accumulators per 16x16 tile: re+ = sum Ar*Br, re- = sum Ai*Bi, im.
  v8f cp00 = {}, cp01 = {}, cp10 = {}, cp11 = {};
  v8f cn00 = {}, cn01 = {}, cn10 = {}, cn11 = {};
  v8f ci00 = {}, ci01 = {}, ci10 = {}, ci11 = {};

  for (int kc = 0; kc < Kdim; kc += BK) {
    // A tile (BM x BK): float4 global loads along K, b128 LDS stores.
#pragma unroll
    for (int i = 0; i < (BM * BK) / (128 * 4); ++i) {
      int idx = t + i * 128;
      int m = idx >> 3;         // 0..63
      int k4 = (idx & 7) * 4;   // 0..28
      size_t g = (size_t)(bm + m) * Kdim + (kc + k4);
      float4 va = *(const float4*)&Ar[g];
      float4 vb = *(const float4*)&Ai[g];
      *(float4*)&sAr[m][k4] = va;
      *(float4*)&sAi[m][k4] = vb;
    }
    // B tile (BK x BN) -> N-major: gather 4 K-rows (each load coalesced in N),
    // one b128 LDS store.
#pragma unroll
    for (int i = 0; i < (BK * BN) / (128 * 4); ++i) {
      int idx = t + i * 128;
      int n = idx & 63;
      int k4 = (idx >> 6) * 4;
      size_t g = (size_t)(kc + k4) * Ndim + (bn + n);
      float4 vr, vi;
      vr.x = Br[g];            vi.x = Bi[g];
      vr.y = Br[g + Ndim];     vi.y = Bi[g + Ndim];
      vr.z = Br[g + 2 * (size_t)Ndim]; vi.z = Bi[g + 2 * (size_t)Ndim];
      vr.w = Br[g + 3 * (size_t)Ndim]; vi.w = Bi[g + 3 * (size_t)Ndim];
      *(float4*)&sBr[n][k4] = vr;
      *(float4*)&sBi[n][k4] = vi;
    }
    if (kc + BK < Kdim) {  // speculative prefetch of next K-chunk cache lines
      __builtin_prefetch(&Ar[(size_t)(bm + (t >> 1)) * Kdim + kc + BK], 0, 0);
      __builtin_prefetch(&Br[(size_t)(kc + BK + (t >> 5)) * Ndim + bn + (t & 31)], 0, 0);
    }
    __syncthreads();

#pragma unroll 4
    for (int k0 = 0; k0 < BK; k0 += 4) {
      const int ka = k0 + half * 2;  // this lane's K pair start (ISA A/B layout)
      v2f ar0 = *(const v2f*)&sAr[wm + l16][ka];
      v2f ai0 = *(const v2f*)&sAi[wm + l16][ka];
      v2f ar1 = *(const v2f*)&sAr[wm + 16 + l16][ka];
      v2f ai1 = *(const v2f*)&sAi[wm + 16 + l16][ka];
      v2f br0 = *(const v2f*)&sBr[wn + l16][ka];
      v2f bi0 = *(const v2f*)&sBi[wn + l16][ka];
      v2f br1 = *(const v2f*)&sBr[wn + 16 + l16][ka];
      v2f bi1 = *(const v2f*)&sBi[wn + 16 + l16][ka];

      cp00 = wmma_f32(ar0, br0, cp00);  cn00 = wmma_f32(ai0, bi0, cn00);
      ci00 = wmma_f32(ar0, bi0, ci00);  ci00 = wmma_f32(ai0, br0, ci00);
      cp01 = wmma_f32(ar0, br1, cp01);  cn01 = wmma_f32(ai0, bi1, cn01);
      ci01 = wmma_f32(ar0, bi1, ci01);  ci01 = wmma_f32(ai0, br1, ci01);
      cp10 = wmma_f32(ar1, br0, cp10);  cn10 = wmma_f32(ai1, bi0, cn10);
      ci10 = wmma_f32(ar1, bi0, ci10);  ci10 = wmma_f32(ai1, br0, ci10);
      cp11 = wmma_f32(ar1, br1, cp11);  cn11 = wmma_f32(ai1, bi1, cn11);
      ci11 = wmma_f32(ar1, bi1, ci11);  ci11 = wmma_f32(ai1, br1, ci11);
    }
    __syncthreads();
  }

  // C layout: VGPR r holds row r (lanes 0-15) / row r+8 (lanes 16-31), col=l16
  const int row0 = bm + wm + half * 8;
  const int col0 = bn + wn + l16;
#pragma unroll
  for (int r = 0; r < 8; ++r) {
#pragma unroll
    for (int mi = 0; mi < 2; ++mi) {
#pragma unroll
      for (int ni = 0; ni < 2; ++ni) {
        float re, im;
        if (mi == 0 && ni == 0) { re = cp00[r] - cn00[r]; im = ci00[r]; }
        else if (mi == 0)       { re = cp01[r] - cn01[r]; im = ci01[r]; }
        else if (ni == 0)       { re = cp10[r] - cn10[r]; im = ci10[r]; }
        else                    { re = cp11[r] - cn11[r]; im = ci11[r]; }
        const int row = row0 + mi * 16 + r;
        const int col = col0 + ni * 16;
        if (interleave) {
          size_t o = ((size_t)row * Ndim + col) * 2;
          Cint[o] = re;
          Cint[o + 1] = im;
        } else {
          size_t o = (size_t)row * Ndim + col;
          Cr[o] = re;
          Ci[o] = im;
        }
      }
    }
  }
}

// A = W - W^H ; M = I - A ; P = I + A ; X0 = alpha * P (= alpha * M^H)
__global__ void build_mp_k(const float* __restrict__ wr, const float* __restrict__ wi,
                           float* __restrict__ Mr, float* __restrict__ Mi,
                           float* __restrict__ Pr, float* __restrict__ Pi,
                           float* __restrict__ Xr, float* __restrict__ Xi,
                           int D, float alpha) {
  int j = blockIdx.x * blockDim.x + threadIdx.x;
  int i = blockIdx.y;
  size_t ij = (size_t)i * D + j;
  size_t ji = (size_t)j * D + i;
  float are = wr[ij] - wr[ji];
  float aim = wi[ij] + wi[ji];
  float dg = (i == j) ? 1.0f : 0.0f;
  Mr[ij] = dg - are;
  Mi[ij] = -aim;
  float pre = dg + are, pim = aim;
  Pr[ij] = pre;
  Pi[ij] = pim;
  Xr[ij] = alpha * pre;
  Xi[ij] = alpha * pim;
}

// T := 2I - T (complex, in place)
__global__ void neg2I_k(float* __restrict__ Tr, float* __restrict__ Ti, int D) {
  size_t idx = (size_t)blockIdx.x * blockDim.x + threadIdx.x;
  int i = (int)(idx / D);
  int j = (int)(idx % D);
  float dg = (i == j) ? 2.0f : 0.0f;
  Tr[idx] = dg - Tr[idx];
  Ti[idx] = -Ti[idx];
}

// Dst = Src^T for both planes (32x32 LDS tiles)
__global__ void transpose2_k(const float* __restrict__ Sr, const float* __restrict__ Si,
                             float* __restrict__ Dr, float* __restrict__ Di, int D) {
  __shared__ float tr[32][33];
  __shared__ float ti[32][33];
  int tx = threadIdx.x & 31;
  int ty = threadIdx.x >> 5;
  int bx = blockIdx.x * 32;
  int by = blockIdx.y * 32;
  for (int r = ty; r < 32; r += 8) {
    size_t g = (size_t)(by + r) * D + bx + tx;
    tr[r][tx] = Sr[g];
    ti[r][tx] = Si[g];
  }
  __syncthreads();
  for (int r = ty; r < 32; r += 8) {
    size_t g = (size_t)(bx + r) * D + by + tx;
    Dr[g] = tr[tx][r];
    Di[g] = ti[tx][r];
  }
}

extern "C" void kernel_launch(void* const* d_in, const int* in_sizes, int n_in,
                              void* d_out, int out_size, void* d_ws, size_t ws_size,
                              hipStream_t stream) {
  const float* x_re = (const float*)d_in[0];
  const float* x_im = (const float*)d_in[1];
  const float* w_re = (const float*)d_in[2];
  const float* w_im = (const float*)d_in[3];
  const int D = 1024;
  const int B = 16384;

  // Workspace: 12 fp32 planes of D*D = 48 MB.
  float* ws = (float*)d_ws;
  const size_t PL = (size_t)D * D;
  float* Mr = ws + 0 * PL;  float* Mi = ws + 1 * PL;
  float* Pr = ws + 2 * PL;  float* Pi = ws + 3 * PL;
  float* Xar = ws + 4 * PL; float* Xai = ws + 5 * PL;
  float* Xbr = ws + 6 * PL; float* Xbi = ws + 7 * PL;
  float* Tr = ws + 8 * PL;  float* Ti = ws + 9 * PL;
  float* Ur = ws + 10 * PL; float* Ui = ws + 11 * PL;

  // alpha = 2/(1+s_max^2) with safety bound ||A||_2 <= 4 (true ~2.6)
  build_mp_k<<<dim3(D / 256, D), 256, 0, stream>>>(w_re, w_im, Mr, Mi, Pr, Pi,
                                                   Xar, Xai, D, 0.111f);

  dim3 gemm_grid(D / BN, D / BM);
  dim3 gemm_block(128);
  float* cr = Xar; float* ci = Xai;
  float* nr = Xbr; float* ni = Xbi;
  for (int it = 0; it < 10; ++it) {  // quadratic convergence; fp32-exact by ~8
    cgemm_nn<<<gemm_grid, gemm_block, 0, stream>>>(Mr, Mi, cr, ci, Tr, Ti,
                                                   nullptr, D, D, D, 0);
    neg2I_k<<<(int)(PL / 256), 256, 0, stream>>>(Tr, Ti, D);
    cgemm_nn<<<gemm_grid, gemm_block, 0, stream>>>(cr, ci, Tr, Ti, nr, ni,
                                                   nullptr, D, D, D, 0);
    float* s;
    s = cr; cr = nr; nr = s;
    s = ci; ci = ni; ni = s;
  }

  // U = M^{-1} * P  -> T ;  then UT = U^T
  cgemm_nn<<<gemm_grid, gemm_block, 0, stream>>>(cr, ci, Pr, Pi, Tr, Ti,
                                                 nullptr, D, D, D, 0);
  transpose2_k<<<dim3(D / 32, D / 32), 256, 0, stream>>>(Tr, Ti, Ur, Ui, D);

  // y = x * U^T, interleaved complex64 directly to d_out
  cgemm_nn<<<dim3(D / BN, B / BM), gemm_block, 0, stream>>>(
      x_re, x_im, Ur, Ui, nullptr, nullptr, (float*)d_out, B, D, D, 1);
}